// AdderModel_18700287607337
// MI455X (gfx1250) — compile-verified
//
#include <hip/hip_runtime.h>
#include <math.h>

// CDNA5 wave32 WMMA operand vectors
typedef __attribute__((ext_vector_type(2))) float v2f;  // A/B for 16x16x4 f32: 2 VGPRs/lane
typedef __attribute__((ext_vector_type(8))) float v8f;  // C/D 16x16 f32: 8 VGPRs/lane

#define TT    64
#define DD    3
#define HDIM  4
#define NPAD  16         // v rows padded to 16 (zeros) so B-operand loads need no selects
#define FFD   2
#define NVOC  10
#define EPSV  1e-6f
#define SCP   (TT + 2)   // score row stride: 264B rows -> 8B aligned, conflict-free f2 access

__global__ __launch_bounds__(64) void tiny_xformer_kernel(
    const int*   __restrict__ idx,
    const float* __restrict__ arcA,
    const float* __restrict__ arcStart,
    const float* __restrict__ arcStride,
    const float* __restrict__ w_ln1,
    const float* __restrict__ w_ln2,
    const float* __restrict__ w_lnf,
    const float* __restrict__ w_qn,
    const float* __restrict__ Wq,
    const float* __restrict__ Wk,
    const float* __restrict__ Wg,
    const float* __restrict__ Wu,
    const float* __restrict__ Wd,
    float*       __restrict__ out)
{
    __shared__ float s_x[TT][DD];
    __shared__ __align__(16) float s_q[TT][HDIM];    // pre-scaled by HD^-0.5
    __shared__ __align__(16) float s_k[TT][HDIM];
    __shared__ __align__(16) float s_vt[NPAD][TT];   // v transposed [dim][token], rows 4..15 = 0
    __shared__ __align__(16) float s_sc[TT][SCP];
    __shared__ float s_o[TT][HDIM];
    __shared__ float s_inv[TT];                      // 1/softmax-sum per row (deferred norm)
    __shared__ float s_tab[NVOC][2];
    __shared__ __align__(16) float s_log[TT * NVOC];

    const int b    = blockIdx.x;
    const int t    = threadIdx.x;        // one thread per token, 64 threads = 2 wave32
    const int lane = threadIdx.x & 31;
    // readfirstlane -> SGPR: compiler emits scalar branches (no EXEC save/restore around WMMA)
    const int wave = __builtin_amdgcn_readfirstlane(threadIdx.x) >> 5;
    const int lh   = lane & 15;
    const bool hi  = lane >= 16;

    const float A    = arcA[0];
    const float ang0 = arcStart[0];
    const float angS = arcStride[0];

    // ---------------- Phase A: embed + ln1 + q/k/v + q-norm + RoPE ----------------
    {
        if (t < NVOC) {                                  // vocab arc table, once per block
            const float av = ang0 + (float)t * angS;
            s_tab[t][0] = A * __cosf(av);
            s_tab[t][1] = A * __sinf(av);
        }
        {   // zero-fill padded v rows 4..15 (contiguous 3072B block) with b128 stores
            float4* zp = (float4*)&s_vt[HDIM][0];        // 192 float4, 64 threads x 3
            const float4 z4 = make_float4(0.f, 0.f, 0.f, 0.f);
            #pragma unroll
            for (int j = 0; j < 3; ++j) zp[j * TT + t] = z4;
        }

        const int tok = idx[(size_t)b * TT + t];
        const float ang = ang0 + (float)tok * angS;
        const float x0 = A * __cosf(ang);
        const float x1 = A * __sinf(ang);
        const float x2 = __sinf((float)t * 1e-4f);       // pe = sin(t * exp(-ln 1e4))

        float ms = (x0*x0 + x1*x1 + x2*x2) * (1.0f / 3.0f);
        float r  = rsqrtf(ms + EPSV);
        const float h0 = x0 * r * w_ln1[0];
        const float h1 = x1 * r * w_ln1[1];
        const float h2 = x2 * r * w_ln1[2];

        float qr[HDIM], kr[HDIM];
        #pragma unroll
        for (int d = 0; d < HDIM; ++d) {
            qr[d] = h0 * Wq[d*DD+0] + h1 * Wq[d*DD+1] + h2 * Wq[d*DD+2];
            kr[d] = h0 * Wk[d*DD+0] + h1 * Wk[d*DD+1] + h2 * Wk[d*DD+2];
        }
        float qms = 0.f, kms = 0.f;
        #pragma unroll
        for (int d = 0; d < HDIM; ++d) { qms += qr[d]*qr[d]; kms += kr[d]*kr[d]; }
        const float qrs = rsqrtf(qms * 0.25f + EPSV);
        const float krs = rsqrtf(kms * 0.25f + EPSV);
        float q[HDIM], k[HDIM];
        #pragma unroll
        for (int d = 0; d < HDIM; ++d) { q[d] = qr[d]*qrs*w_qn[d]; k[d] = kr[d]*krs*w_qn[d]; }

        // RoPE, theta = 3.0, HD = 4 -> inv_freq = {1, 3^-0.5}
        const float f0 = (float)t;
        const float f1 = (float)t * 0.5773502691896258f;
        const float c0 = __cosf(f0), sn0 = __sinf(f0);
        const float c1 = __cosf(f1), sn1 = __sinf(f1);

        s_x[t][0] = x0; s_x[t][1] = x1; s_x[t][2] = x2;
        // q pre-scaled by HD^-0.5 = 0.5 (linear, commutes with rope)
        s_q[t][0] = 0.5f*(q[0]*c0 - q[1]*sn0);  s_q[t][1] = 0.5f*(q[0]*sn0 + q[1]*c0);
        s_q[t][2] = 0.5f*(q[2]*c1 - q[3]*sn1);  s_q[t][3] = 0.5f*(q[2]*sn1 + q[3]*c1);
        s_k[t][0] = k[0]*c0 - k[1]*sn0;  s_k[t][1] = k[0]*sn0 + k[1]*c0;
        s_k[t][2] = k[2]*c1 - k[3]*sn1;  s_k[t][3] = k[2]*sn1 + k[3]*c1;
        s_vt[0][t] = kr[0]; s_vt[1][t] = kr[1];          // v = h@Wk^T, transposed
        s_vt[2][t] = kr[2]; s_vt[3][t] = kr[3];
    }
    __syncthreads();

    // ------- Phase B: scores = q @ k^T  (V_WMMA_F32_16X16X4_F32, causal block skipping) --
    // A 16x4: lanes0-15 K=0/1, lanes16-31 K=2/3 -> single 8B-aligned ds_load_b64 per lane.
    for (int rb = wave; rb < 4; rb += 2) {              // SGPR loop: scalar branches only
        const int rrow = rb * 16 + lh;
        const v2f a = *(const v2f*)&s_q[rrow][hi ? 2 : 0];
        #pragma unroll
        for (int cb = 0; cb < 4; ++cb) {
            const int n = cb * 16 + lh;
            if (cb <= rb) {                              // at/below diagonal: compute
                const v2f bm = *(const v2f*)&s_k[n][hi ? 2 : 0];
                v8f c = {};
                c = __builtin_amdgcn_wmma_f32_16x16x4_f32(false, a, false, bm,
                                                          (short)0, c, false, false);
                if (cb < rb) {                           // strictly below: no mask
                    #pragma unroll
                    for (int i = 0; i < 8; ++i)
                        s_sc[rb*16 + i + (hi ? 8 : 0)][n] = c[i];
                } else {                                 // diagonal tile: per-element mask
                    #pragma unroll
                    for (int i = 0; i < 8; ++i) {
                        const int row = rb*16 + i + (hi ? 8 : 0);
                        s_sc[row][n] = (n > row) ? -INFINITY : c[i];
                    }
                }
            } else {                                     // above diagonal: fully masked
                #pragma unroll
                for (int i = 0; i < 8; ++i)
                    s_sc[rb*16 + i + (hi ? 8 : 0)][n] = -INFINITY;
            }
        }
    }
    __syncthreads();

    // -------- Phase C: row max + unnormalized exp; 1/sum deferred to Phase E -------------
    {
        v2f* rowp = (v2f*)&s_sc[t][0];                   // 264B row stride -> 8B aligned
        float m = -INFINITY;
        #pragma unroll
        for (int j = 0; j < TT/2; ++j) {
            const v2f p = rowp[j];
            m = fmaxf(m, fmaxf(p[0], p[1]));
        }
        float s = 0.f;
        #pragma unroll
        for (int j = 0; j < TT/2; ++j) {
            v2f p = rowp[j];
            p[0] = __expf(p[0] - m);                     // masked cols -> exp(-inf) = 0
            p[1] = __expf(p[1] - m);
            rowp[j] = p;
            s += p[0] + p[1];
        }
        s_inv[t] = 1.0f / s;                             // softmax norm commutes with attn@v
    }
    __syncthreads();

    // ---------------- Phase D: out = e @ v  (chained K=4 WMMAs, N padded to 16) ----------
    for (int rb = wave; rb < 4; rb += 2) {
        v8f c = {};
        const int rrow = rb * 16 + lh;
        const int n = lh;
        #pragma unroll
        for (int kc = 0; kc < 16; ++kc) {
            const v2f a  = *(const v2f*)&s_sc[rrow][kc*4 + (hi ? 2 : 0)];   // 8B aligned
            const v2f bm = *(const v2f*)&s_vt[n][kc*4 + (hi ? 2 : 0)];      // rows>=4 are zero
            c = __builtin_amdgcn_wmma_f32_16x16x4_f32(false, a, false, bm,
                                                      (short)0, c, false, false);
        }
        if (n < HDIM) {
            #pragma unroll
            for (int i = 0; i < 8; ++i)
                s_o[rb*16 + i + (hi ? 8 : 0)][n] = c[i];
        }
    }
    __syncthreads();

    // ---------------- Phase E: normalize + residual + MLP + final norm + logits ----------
    {
        const float inv = s_inv[t];                      // apply deferred softmax 1/sum
        const float o0 = s_o[t][0] * inv, o1 = s_o[t][1] * inv;
        const float o2 = s_o[t][2] * inv, o3 = s_o[t][3] * inv;
        float x0 = s_x[t][0], x1 = s_x[t][1], x2 = s_x[t][2];
        // x += out @ Wq   (Wq: HD x D row-major)
        x0 += o0*Wq[0*DD+0] + o1*Wq[1*DD+0] + o2*Wq[2*DD+0] + o3*Wq[3*DD+0];
        x1 += o0*Wq[0*DD+1] + o1*Wq[1*DD+1] + o2*Wq[2*DD+1] + o3*Wq[3*DD+1];
        x2 += o0*Wq[0*DD+2] + o1*Wq[1*DD+2] + o2*Wq[2*DD+2] + o3*Wq[3*DD+2];

        float ms = (x0*x0 + x1*x1 + x2*x2) * (1.0f / 3.0f);
        float r  = rsqrtf(ms + EPSV);
        const float h0 = x0*r*w_ln2[0], h1 = x1*r*w_ln2[1], h2 = x2*r*w_ln2[2];

        float y[FFD];
        #pragma unroll
        for (int f = 0; f < FFD; ++f) {
            const float g = h0*Wg[f*DD+0] + h1*Wg[f*DD+1] + h2*Wg[f*DD+2];
            const float u = h0*Wu[f*DD+0] + h1*Wu[f*DD+1] + h2*Wu[f*DD+2];
            y[f] = (g / (1.0f + __expf(-g))) * u;        // silu(g)*u
        }
        x0 += y[0]*Wd[0*FFD+0] + y[1]*Wd[0*FFD+1];
        x1 += y[0]*Wd[1*FFD+0] + y[1]*Wd[1*FFD+1];
        x2 += y[0]*Wd[2*FFD+0] + y[1]*Wd[2*FFD+1];

        ms = (x0*x0 + x1*x1 + x2*x2) * (1.0f / 3.0f);
        r  = rsqrtf(ms + EPSV);
        x0 = x0 * r * w_lnf[0];
        x1 = x1 * r * w_lnf[1];

        #pragma unroll
        for (int vv = 0; vv < NVOC; ++vv)
            s_log[t * NVOC + vv] = x0 * s_tab[vv][0] + x1 * s_tab[vv][1];
    }
    __syncthreads();

    // ---------------- Phase F: coalesced logits store (float2, 256B/wave/iter) -----------
    {
        const float2* sl = (const float2*)s_log;
        float2* op = (float2*)(out + (size_t)b * TT * NVOC);
        #pragma unroll
        for (int j = 0; j < (TT * NVOC) / (2 * TT); ++j) {   // 5 iterations
            const int l = j * TT + t;
            op[l] = sl[l];
        }
    }
}

extern "C" void kernel_launch(void* const* d_in, const int* in_sizes, int n_in,
                              void* d_out, int out_size, void* d_ws, size_t ws_size,
                              hipStream_t stream) {
    const int*   idx       = (const int*)  d_in[0];
    const float* arcA      = (const float*)d_in[1];
    const float* arcStart  = (const float*)d_in[2];
    const float* arcStride = (const float*)d_in[3];
    const float* w_ln1     = (const float*)d_in[4];
    const float* w_ln2     = (const float*)d_in[5];
    const float* w_lnf     = (const float*)d_in[6];
    const float* w_qn      = (const float*)d_in[7];
    const float* Wq        = (const float*)d_in[8];
    const float* Wk        = (const float*)d_in[9];
    const float* Wg        = (const float*)d_in[10];
    const float* Wu        = (const float*)d_in[11];
    const float* Wd        = (const float*)d_in[12];
    float*       out       = (float*)d_out;

    const int nBatch = in_sizes[0] / TT;   // 16384
    tiny_xformer_kernel<<<dim3(nBatch), dim3(TT), 0, stream>>>(
        idx, arcA, arcStart, arcStride, w_ln1, w_ln2, w_lnf, w_qn,
        Wq, Wk, Wg, Wu, Wd, out);
}